// AttentionBlock_12360915878293
// MI455X (gfx1250) — compile-verified
//
#include <hip/hip_runtime.h>
#include <hip/hip_bf16.h>
#include <math.h>

// ---------------------------------------------------------------------------
// CDNA5 (gfx1250) wave32 WMMA f32 16x16x4 helpers.
// A 16x4: lanes 0-15 -> M=lane, K={k0,k0+1}; lanes 16-31 -> M=lane-16, K={k0+2,k0+3}
// B 4x16: lanes 0-15 -> N=lane, K={k0,k0+1}; lanes 16-31 -> N=lane-16, K={k0+2,k0+3}
// C/D 16x16: VGPR v, lanes 0-15 -> M=v,  N=lane; lanes 16-31 -> M=v+8, N=lane-16
// ---------------------------------------------------------------------------
typedef __attribute__((ext_vector_type(2))) float v2f;
typedef __attribute__((ext_vector_type(8))) float v8f;

#define WMMA_F32(a, b, c) \
  __builtin_amdgcn_wmma_f32_16x16x4_f32(false, (a), false, (b), (short)0, (c), false, false)

// gfx1250 async copy: 16B/lane global -> LDS, tracked by ASYNCcnt (ISA 10.x / 15.18.3)
static __device__ __forceinline__ void async_load_b128(unsigned lds_off, const float* gaddr) {
  asm volatile("global_load_async_to_lds_b128 %0, %1, off" ::"v"(lds_off), "v"(gaddr)
               : "memory");
}
// async loads complete in order -> s_wait_asynccnt N releases all but the last N issues
template <int N>
static __device__ __forceinline__ void wait_async_le() {
  asm volatile("s_wait_asynccnt %0" ::"i"(N) : "memory");
}
static __device__ __forceinline__ unsigned lds_off_of(const void* p) {
  return (unsigned)(unsigned long long)p;  // LDS generic addr: offset lives in addr[31:0]
}

// A-fragment from a row-major matrix (also serves as "transposed-B" fragment:
// B[k][n] = M[n][k] has identical addressing with row0 = col0 of the tile).
static __device__ __forceinline__ v2f frag_a(const float* M, int ld, int row0, int k0, int lane) {
  int r = lane & 15, kb = (lane >> 4) * 2;
  v2f a;
  a.x = M[(row0 + r) * ld + k0 + kb];
  a.y = M[(row0 + r) * ld + k0 + kb + 1];
  return a;
}

// B-fragment from a row-major matrix B[k][n]
static __device__ __forceinline__ v2f frag_b(const float* M, int ld, int k0, int col0, int lane) {
  int n = lane & 15, kb = (lane >> 4) * 2;
  v2f b;
  b.x = M[(k0 + kb) * ld + col0 + n];
  b.y = M[(k0 + kb + 1) * ld + col0 + n];
  return b;
}

static __device__ __forceinline__ void store_tile(float* M, int ld, int row0, int col0,
                                                  int lane, v8f c) {
  int half = lane >> 4, col = lane & 15;
#pragma unroll
  for (int v = 0; v < 8; ++v) M[(row0 + v + half * 8) * ld + col0 + col] = c[v];
}

// 64x64x64 matmul tile from padded-LDS (ld=65) operands
static __device__ __forceinline__ v8f mm64(const float* As, const float* Bs, int mt, int nt,
                                           int lane) {
  v8f c = {};
  int row0 = mt * 16, col0 = nt * 16;
#pragma unroll
  for (int k0 = 0; k0 < 64; k0 += 4) {
    v2f a = frag_a(As, 65, row0, k0, lane);
    v2f b = frag_b(Bs, 65, k0, col0, lane);
    c = WMMA_F32(a, b, c);
  }
  return c;
}

#define BATCH 128
#define CIN 256
#define HWN 1024
#define ATT 64
#define TRI 2080
#define PLANES 256
#define BN_EPS 1e-5f

// ---------------------------------------------------------------------------
// K1: act[b][64][1024] = relu(bn(conv_w @ x[b]))   (M=64, N=1024 tiled by 128, K=256)
// grid (8, 128) x 256 threads. Double-buffered async-b128 LDS staging: chunk
// kc+1's DMA overlaps chunk kc's WMMAs (async loads retire in order, so
// s_wait_asynccnt 6 releases exactly the previous chunk).
// xs ld=136: fragment banks (8k+n)%64 distinct across all 32 lanes.
// wlds ld=36: A-fragment banks (36r+k)%64 distinct across the 16 M-lanes.
// ---------------------------------------------------------------------------
__global__ void conv_bn_relu_kernel(const float* __restrict__ x, const float* __restrict__ cw,
                                    const float* __restrict__ g, const float* __restrict__ be,
                                    const float* __restrict__ mn, const float* __restrict__ vr,
                                    float* __restrict__ act) {
  __shared__ float xs[2][32 * 136];    // 32 x 128 x-panels, padded
  __shared__ float wlds[2][64 * 36];   // 64 x 32 conv_w panels, padded
  int b = blockIdx.y;
  int n0 = blockIdx.x * 128;
  int t = threadIdx.x, lane = t & 31, w = t >> 5;
  int mt = w >> 1;
  int ntb = (w & 1) * 4;
  int row0 = mt * 16;
  const float* xb = x + (size_t)b * CIN * HWN;

  // stage one 64x32 conv_w panel + one 32x128 x panel (6 async issues/thread)
  auto stage = [&](int kc, int buf) {
    int k0 = kc * 32;
#pragma unroll
    for (int j = 0; j < 2; ++j) {
      int q = w * 64 + j * 32 + lane;
      int r = q >> 3, c4 = (q & 7) * 4;
      async_load_b128(lds_off_of(&wlds[buf][r * 36 + c4]), cw + r * CIN + k0 + c4);
    }
#pragma unroll
    for (int j = 0; j < 4; ++j) {
      int q = w * 128 + j * 32 + lane;
      int kr = q >> 5, c4 = (q & 31) * 4;
      async_load_b128(lds_off_of(&xs[buf][kr * 136 + c4]), xb + (k0 + kr) * HWN + n0 + c4);
    }
  };

  v8f a0 = {}, a1 = {}, a2 = {}, a3 = {};
  stage(0, 0);
  for (int kc = 0; kc < 8; ++kc) {
    int buf = kc & 1;
    if (kc < 7) {
      stage(kc + 1, buf ^ 1);   // prefetch next panel into other buffer
      wait_async_le<6>();       // previous panel landed; next may stay in flight
    } else {
      wait_async_le<0>();
    }
    __syncthreads();
#pragma unroll
    for (int kk = 0; kk < 32; kk += 4) {
      v2f af = frag_a(wlds[buf], 36, row0, kk, lane);
      v2f b0 = frag_b(xs[buf], 136, kk, (ntb + 0) * 16, lane);
      v2f b1 = frag_b(xs[buf], 136, kk, (ntb + 1) * 16, lane);
      v2f b2 = frag_b(xs[buf], 136, kk, (ntb + 2) * 16, lane);
      v2f b3 = frag_b(xs[buf], 136, kk, (ntb + 3) * 16, lane);
      a0 = WMMA_F32(af, b0, a0);
      a1 = WMMA_F32(af, b1, a1);
      a2 = WMMA_F32(af, b2, a2);
      a3 = WMMA_F32(af, b3, a3);
    }
    __syncthreads();  // release buf for the staging two iterations ahead
  }

  int half = lane >> 4, cc = lane & 15;
  float* actb = act + (size_t)b * ATT * HWN;
  v8f accs[4] = {a0, a1, a2, a3};
#pragma unroll
  for (int v = 0; v < 8; ++v) {
    int row = row0 + v + half * 8;
    float s = g[row] * rsqrtf(vr[row] + BN_EPS);
    float sh = be[row] - mn[row] * s;
#pragma unroll
    for (int j = 0; j < 4; ++j) {
      int col = n0 + (ntb + j) * 16 + cc;
      float val = accs[j][v] * s + sh;
      actb[row * HWN + col] = val > 0.0f ? val : 0.0f;
    }
  }
}

// ---------------------------------------------------------------------------
// K2: cov[b] = act act^T / M - mu mu^T   (64x64, K=1024 in 16 async-staged panels)
// one WG per batch; wave w owns tiles w and w+8 of the 4x4 tile grid.
// sm ld=68 (rows 16B-aligned for b128; fragment banks (4r+k)%64 conflict-free).
// Double-buffered async staging as in K1 (4 async issues/thread per panel).
// ---------------------------------------------------------------------------
__global__ void covpool_kernel(const float* __restrict__ act, float* __restrict__ cov) {
  __shared__ float sm[2][ATT * 68];
  __shared__ float spart[ATT * 4];
  __shared__ float smu[ATT];
  int b = blockIdx.x;
  int t = threadIdx.x, lane = t & 31, w = t >> 5;
  const float* ab = act + (size_t)b * ATT * HWN;

  // row means
  {
    int row = t >> 2, part = t & 3;
    float s = 0.0f;
    const float* p = ab + row * HWN + part * 256;
    for (int i = 0; i < 256; ++i) s += p[i];
    spart[row * 4 + part] = s;
  }
  __syncthreads();
  if (t < ATT)
    smu[t] = (spart[t * 4] + spart[t * 4 + 1] + spart[t * 4 + 2] + spart[t * 4 + 3]) *
             (1.0f / (float)HWN);
  __syncthreads();

  auto stage = [&](int p, int buf) {
#pragma unroll
    for (int j = 0; j < 4; ++j) {
      int q = w * 128 + j * 32 + lane;
      int r = q >> 4, c4 = (q & 15) * 4;
      async_load_b128(lds_off_of(&sm[buf][r * 68 + c4]), ab + r * HWN + p * 64 + c4);
    }
  };

  int mt1 = w >> 2, mt2 = mt1 + 2, nt = w & 3;
  v8f acc1 = {}, acc2 = {};

  stage(0, 0);
  for (int p = 0; p < 16; ++p) {
    int buf = p & 1;
    if (p < 15) {
      stage(p + 1, buf ^ 1);
      wait_async_le<4>();
    } else {
      wait_async_le<0>();
    }
    __syncthreads();
#pragma unroll
    for (int k0 = 0; k0 < 64; k0 += 4) {
      v2f fm1 = frag_a(sm[buf], 68, mt1 * 16, k0, lane);
      v2f fm2 = frag_a(sm[buf], 68, mt2 * 16, k0, lane);
      v2f fn = frag_a(sm[buf], 68, nt * 16, k0, lane);  // B[k][n]=act[n][k]: transposed==A-style
      acc1 = WMMA_F32(fm1, fn, acc1);
      acc2 = WMMA_F32(fm2, fn, acc2);
    }
    __syncthreads();
  }

  int half = lane >> 4, cc = lane & 15;
  float* cb = cov + (size_t)b * ATT * ATT;
  const float invM = 1.0f / (float)HWN;
#pragma unroll
  for (int v = 0; v < 8; ++v) {
    int r1 = mt1 * 16 + v + half * 8;
    int r2 = mt2 * 16 + v + half * 8;
    int col = nt * 16 + cc;
    cb[r1 * ATT + col] = acc1[v] * invM - smu[r1] * smu[col];
    cb[r2 * ATT + col] = acc2[v] * invM - smu[r2] * smu[col];
  }
}

// ---------------------------------------------------------------------------
// K3: Newton-Schulz sqrt (iterN=5) on 64x64 + fused triuvec. One WG per batch.
// 3 LDS buffers (ld=65), 12 WMMA matmuls, register ping-pong for in-place updates.
// ---------------------------------------------------------------------------
__global__ void ns_sqrt_kernel(const float* __restrict__ cov, float* __restrict__ vec) {
  __shared__ float B0[ATT * 65];  // An / T / W
  __shared__ float B1[ATT * 65];  // T0 -> Z
  __shared__ float B2[ATT * 65];  // Y
  __shared__ float sdiag[ATT];
  __shared__ float s_norm;
  int b = blockIdx.x;
  int t = threadIdx.x, lane = t & 31, w = t >> 5;
  const float* A = cov + (size_t)b * ATT * ATT;

  if (t < ATT) sdiag[t] = A[t * (ATT + 1)];
  __syncthreads();
  if (t == 0) {
    float s = 0.0f;
    for (int i = 0; i < ATT; ++i) s += sdiag[i];
    s_norm = s;
  }
  __syncthreads();
  float normA = s_norm;
  float inv = 1.0f / normA;

#pragma unroll
  for (int i = 0; i < 16; ++i) {
    int e = t * 16 + i;
    int r = e >> 6, k = e & 63;
    float an = A[e] * inv;
    B0[r * 65 + k] = an;                                      // An
    B1[r * 65 + k] = 0.5f * ((r == k ? 3.0f : 0.0f) - an);    // T0 = 0.5*(3I - An)
  }
  __syncthreads();

  int mt1 = w >> 2, mt2 = mt1 + 2, nt = w & 3;
  int half = lane >> 4, cc = lane & 15;
  int col = nt * 16 + cc;

  // Y0 = An @ T0  -> B2 ; Z0 = T0 stays in B1
  {
    v8f y1 = mm64(B0, B1, mt1, nt, lane);
    v8f y2 = mm64(B0, B1, mt2, nt, lane);
    store_tile(B2, 65, mt1 * 16, nt * 16, lane, y1);
    store_tile(B2, 65, mt2 * 16, nt * 16, lane, y2);
  }
  __syncthreads();

  for (int it = 0; it < 3; ++it) {  // range(1, iterN-1)
    // T = 0.5*(3I - Z@Y) -> B0
    v8f zy1 = mm64(B1, B2, mt1, nt, lane);
    v8f zy2 = mm64(B1, B2, mt2, nt, lane);
#pragma unroll
    for (int v = 0; v < 8; ++v) {
      int r1 = mt1 * 16 + v + half * 8;
      int r2 = mt2 * 16 + v + half * 8;
      zy1[v] = 0.5f * ((r1 == col ? 3.0f : 0.0f) - zy1[v]);
      zy2[v] = 0.5f * ((r2 == col ? 3.0f : 0.0f) - zy2[v]);
    }
    __syncthreads();  // all waves done reading previous B0 before overwrite
    store_tile(B0, 65, mt1 * 16, nt * 16, lane, zy1);
    store_tile(B0, 65, mt2 * 16, nt * 16, lane, zy2);
    __syncthreads();
    // Ynew = Y@T, Znew = T@Z (held in regs, then committed)
    v8f yn1 = mm64(B2, B0, mt1, nt, lane);
    v8f yn2 = mm64(B2, B0, mt2, nt, lane);
    v8f zn1 = mm64(B0, B1, mt1, nt, lane);
    v8f zn2 = mm64(B0, B1, mt2, nt, lane);
    __syncthreads();
    store_tile(B2, 65, mt1 * 16, nt * 16, lane, yn1);
    store_tile(B2, 65, mt2 * 16, nt * 16, lane, yn2);
    store_tile(B1, 65, mt1 * 16, nt * 16, lane, zn1);
    store_tile(B1, 65, mt2 * 16, nt * 16, lane, zn2);
    __syncthreads();
  }

  // final: R = 0.5 * Y @ (3I - Z@Y) * sqrt(normA), triuvec'd
  {
    v8f w1 = mm64(B1, B2, mt1, nt, lane);
    v8f w2 = mm64(B1, B2, mt2, nt, lane);
#pragma unroll
    for (int v = 0; v < 8; ++v) {
      int r1 = mt1 * 16 + v + half * 8;
      int r2 = mt2 * 16 + v + half * 8;
      w1[v] = (r1 == col ? 3.0f : 0.0f) - w1[v];
      w2[v] = (r2 == col ? 3.0f : 0.0f) - w2[v];
    }
    __syncthreads();
    store_tile(B0, 65, mt1 * 16, nt * 16, lane, w1);
    store_tile(B0, 65, mt2 * 16, nt * 16, lane, w2);
    __syncthreads();
    v8f r1 = mm64(B2, B0, mt1, nt, lane);
    v8f r2 = mm64(B2, B0, mt2, nt, lane);
    float sc = 0.5f * sqrtf(normA);
    float* vb = vec + (size_t)b * TRI;
#pragma unroll
    for (int v = 0; v < 8; ++v) {
      int row = mt1 * 16 + v + half * 8;
      if (col >= row) vb[row * ATT - (row * (row - 1)) / 2 + (col - row)] = r1[v] * sc;
      row = mt2 * 16 + v + half * 8;
      if (col >= row) vb[row * ATT - (row * (row - 1)) / 2 + (col - row)] = r2[v] * sc;
    }
  }
}

// ---------------------------------------------------------------------------
// K4: gate = sigmoid(vec @ fc_w^T + fc_b)  (M=128, N=256, K=2080) — 1 tile/wave
// ---------------------------------------------------------------------------
__global__ void fc_sigmoid_kernel(const float* __restrict__ vec, const float* __restrict__ fcw,
                                  const float* __restrict__ fcb, float* __restrict__ gate) {
  int t = threadIdx.x, lane = t & 31, w = t >> 5;
  int tile = blockIdx.x * 8 + w;  // 0..127
  int mt = tile >> 4, nt = tile & 15;
  int row0 = mt * 16, col0 = nt * 16;
  v8f c = {};
  for (int k0 = 0; k0 < TRI; k0 += 4) {
    v2f a = frag_a(vec, TRI, row0, k0, lane);
    v2f bb = frag_a(fcw, TRI, col0, k0, lane);  // B[k][n] = fc_w[n][k]: transposed == A-style
    c = WMMA_F32(a, bb, c);
  }
  int half = lane >> 4, cc = lane & 15;
  int col = col0 + cc;
  float bias = fcb[col];
#pragma unroll
  for (int v = 0; v < 8; ++v) {
    int row = row0 + v + half * 8;
    float val = c[v] + bias;
    gate[row * PLANES + col] = 1.0f / (1.0f + __expf(-val));
  }
}

// ---------------------------------------------------------------------------
// K5: out = x * gate[b, c] broadcast, float4 vectorized (HW=1024 % 4 == 0)
// ---------------------------------------------------------------------------
__global__ void scale_kernel(const float* __restrict__ x, const float* __restrict__ gate,
                             float* __restrict__ out) {
  size_t i = (size_t)blockIdx.x * blockDim.x + threadIdx.x;
  if (i >= (size_t)BATCH * CIN * HWN / 4) return;
  size_t e = i * 4;
  int b = (int)(e >> 18);          // / (256*1024)
  int c = (int)((e >> 10) & 255);  // / 1024 % 256
  float g = gate[b * PLANES + c];
  float4 v = ((const float4*)x)[i];
  v.x *= g; v.y *= g; v.z *= g; v.w *= g;
  ((float4*)out)[i] = v;
}

// ---------------------------------------------------------------------------
extern "C" void kernel_launch(void* const* d_in, const int* in_sizes, int n_in,
                              void* d_out, int out_size, void* d_ws, size_t ws_size,
                              hipStream_t stream) {
  const float* x = (const float*)d_in[0];
  const float* conv_w = (const float*)d_in[1];
  const float* bn_g = (const float*)d_in[2];
  const float* bn_b = (const float*)d_in[3];
  const float* bn_m = (const float*)d_in[4];
  const float* bn_v = (const float*)d_in[5];
  const float* fc_w = (const float*)d_in[6];
  const float* fc_b = (const float*)d_in[7];
  float* out = (float*)d_out;

  char* ws = (char*)d_ws;
  float* act = (float*)ws;                                       // 128*64*1024*4 = 32 MiB
  float* cov = (float*)(ws + (size_t)33554432);                  // 128*64*64*4   =  2 MiB
  float* vec = (float*)(ws + (size_t)33554432 + 2097152);        // 128*2080*4    ~  1 MiB
  float* gate = (float*)(ws + (size_t)33554432 + 2097152 + 1064960);  // 128*256*4

  conv_bn_relu_kernel<<<dim3(8, BATCH), 256, 0, stream>>>(x, conv_w, bn_g, bn_b, bn_m, bn_v, act);
  covpool_kernel<<<BATCH, 256, 0, stream>>>(act, cov);
  ns_sqrt_kernel<<<BATCH, 256, 0, stream>>>(cov, vec);
  fc_sigmoid_kernel<<<16, 256, 0, stream>>>(vec, fc_w, fc_b, gate);
  scale_kernel<<<(BATCH * CIN * HWN / 4 + 255) / 256, 256, 0, stream>>>(x, gate, out);
}